// DescriptorMatcher_62835371540574
// MI455X (gfx1250) — compile-verified
//
#include <hip/hip_runtime.h>
#include <hip/hip_bf16.h>
#include <math.h>

typedef __attribute__((ext_vector_type(2))) float v2f;
typedef __attribute__((ext_vector_type(8))) float v8f;

#define D 128
#define NK 32           // K-steps of 4 per 128-deep dot product
#define D1_STRIDE 132   // pad strip rows -> bank step 4, conflict free
#define EPSQ 1e-12f

// ---------------------------------------------------------------------------
// Kernel 1: row squared-norms for both descriptor matrices into workspace.
// ---------------------------------------------------------------------------
__global__ void dm_norms_kernel(const float* __restrict__ d1,
                                const float* __restrict__ d2,
                                float* __restrict__ sq, int B1, int B2) {
    int i = blockIdx.x * blockDim.x + threadIdx.x;
    int total = B1 + B2;
    if (i >= total) return;
    const float* row = (i < B1) ? (d1 + (size_t)i * D) : (d2 + (size_t)(i - B1) * D);
    float s = 0.f;
#pragma unroll 8
    for (int k = 0; k < D; k += 4) {
        float4 v = *(const float4*)(row + k);
        s += v.x * v.x + v.y * v.y + v.z * v.z + v.w * v.w;
    }
    sq[i] = s;
}

// ---------------------------------------------------------------------------
// Helpers for the match kernel
// ---------------------------------------------------------------------------
__device__ __forceinline__ void load_btile(const float* __restrict__ d2,
                                           const float* __restrict__ sq2g,
                                           int col, int koff,
                                           v2f (&b)[NK], float& sq2v) {
    const float* d2row = d2 + (size_t)col * D;
    sq2v = sq2g[col];
#pragma unroll
    for (int i = 0; i < NK; ++i)
        b[i] = *(const v2f*)&d2row[4 * i + koff];   // B[k0..k0+1][n=li]
}

// Two independent WMMA chains (rows 0-15 and 16-31) sharing one B tile.
__device__ __forceinline__ void do_tile(const v2f (&a0)[NK], const v2f (&a1)[NK],
                                        const v2f (&b)[NK],
                                        const float (&sq1loc)[16], float sq2v, int col,
                                        float (&best2)[16], int (&bidx)[16]) {
    v8f acc0 = {};
    v8f acc1 = {};
#pragma unroll
    for (int i = 0; i < NK; ++i) {
        acc0 = __builtin_amdgcn_wmma_f32_16x16x4_f32(
            false, a0[i], false, b[i], (short)0, acc0, false, false);
        acc1 = __builtin_amdgcn_wmma_f32_16x16x4_f32(
            false, a1[i], false, b[i], (short)0, acc1, false, false);
    }
#pragma unroll
    for (int r = 0; r < 8; ++r) {
        float d2v = sq1loc[r] + sq2v - 2.0f * acc0[r];
        if (d2v < best2[r] || (d2v == best2[r] && col < bidx[r])) {
            best2[r] = d2v;
            bidx[r]  = col;
        }
    }
#pragma unroll
    for (int r = 0; r < 8; ++r) {
        float d2v = sq1loc[8 + r] + sq2v - 2.0f * acc1[r];
        if (d2v < best2[8 + r] || (d2v == best2[8 + r] && col < bidx[8 + r])) {
            best2[8 + r] = d2v;
            bidx[8 + r]  = col;
        }
    }
}

// ---------------------------------------------------------------------------
// Kernel 2: per block -> 32 rows of desc1 (two 16-row strips). 8 waves sweep
// 16-col tiles of desc2; each B register tile feeds two WMMA chains.
// ---------------------------------------------------------------------------
__global__ __launch_bounds__(256)
void dm_match_kernel(const float* __restrict__ d1,
                     const float* __restrict__ d2,
                     const float* __restrict__ sq1g,
                     const float* __restrict__ sq2g,
                     float* __restrict__ out_d,
                     int* __restrict__ out_i,
                     int B2) {
    __shared__ __align__(16) float d1s[32 * D1_STRIDE];
    __shared__ float wmin[8][32];
    __shared__ int   widx[8][32];

    const int t    = threadIdx.x;
    const int lane = t & 31;
    const int wave = t >> 5;
    const int li   = lane & 15;   // column lane within tile / A-frag row
    const int hi   = lane >> 4;   // which K-half of the 4-wide K block
    const int koff = hi * 2;
    const int row0 = blockIdx.x * 32;

    // Stage the 32x128 desc1 strip into padded LDS (4 x float4 per thread).
    {
        int r = t >> 3;          // 0..31
        int c = (t & 7) * 16;    // 0,16,...,112
        const float4* src = (const float4*)(d1 + (size_t)(row0 + r) * D + c);
        float4 a0 = src[0], a1 = src[1], a2 = src[2], a3 = src[3];
        float* dst = &d1s[r * D1_STRIDE + c];
        *(float4*)(dst + 0)  = a0;
        *(float4*)(dst + 4)  = a1;
        *(float4*)(dst + 8)  = a2;
        *(float4*)(dst + 12) = a3;
    }
    __syncthreads();

    // Hoist A fragments for both strips into registers (128 VGPRs).
    v2f afrag0[NK], afrag1[NK];
#pragma unroll
    for (int i = 0; i < NK; ++i) {
        afrag0[i] = *(const v2f*)&d1s[li * D1_STRIDE + 4 * i + koff];
        afrag1[i] = *(const v2f*)&d1s[(16 + li) * D1_STRIDE + 4 * i + koff];
    }

    // Lane's accumulator rows: strip0 -> m = r + 8*hi; strip1 -> 16 + same.
    float sq1loc[16];
#pragma unroll
    for (int r = 0; r < 8; ++r) {
        sq1loc[r]     = sq1g[row0 + r + 8 * hi];
        sq1loc[8 + r] = sq1g[row0 + 16 + r + 8 * hi];
    }

    float best2[16];
    int   bidx[16];
#pragma unroll
    for (int r = 0; r < 16; ++r) { best2[r] = INFINITY; bidx[r] = 0x7fffffff; }

    // Tile j for this wave covers columns colbase + 128*j .. +15.
    const int ntiles  = B2 >> 4;
    const int nj      = ntiles >> 3;          // tiles per wave
    const int colbase = wave * 16 + li;

    v2f b0[NK], b1[NK];
    float s0, s1;

    load_btile(d2, sq2g, colbase, koff, b0, s0);
    int j = 0;
    for (; j + 1 < nj; j += 2) {
        load_btile(d2, sq2g, colbase + 128 * (j + 1), koff, b1, s1);
        do_tile(afrag0, afrag1, b0, sq1loc, s0, colbase + 128 * j, best2, bidx);
        if (j + 2 < nj)
            load_btile(d2, sq2g, colbase + 128 * (j + 2), koff, b0, s0);
        do_tile(afrag0, afrag1, b1, sq1loc, s1, colbase + 128 * (j + 1), best2, bidx);
    }
    if (j < nj)   // odd tail (also handles nj==1)
        do_tile(afrag0, afrag1, b0, sq1loc, s0, colbase + 128 * j, best2, bidx);

    // Fold the 16 column-lanes per row half (xor masks < 16 stay in-half).
#pragma unroll
    for (int m = 8; m >= 1; m >>= 1) {
#pragma unroll
        for (int r = 0; r < 16; ++r) {
            float od = __shfl_xor(best2[r], m, 32);
            int   oi = __shfl_xor(bidx[r], m, 32);
            if (od < best2[r] || (od == best2[r] && oi < bidx[r])) {
                best2[r] = od;
                bidx[r]  = oi;
            }
        }
    }

    if (li == 0) {
#pragma unroll
        for (int r = 0; r < 8; ++r) {
            wmin[wave][r + 8 * hi]      = best2[r];
            widx[wave][r + 8 * hi]      = bidx[r];
            wmin[wave][16 + r + 8 * hi] = best2[8 + r];
            widx[wave][16 + r + 8 * hi] = bidx[8 + r];
        }
    }
    __syncthreads();

    // Final fold across the 8 waves; threads 0..31 each own one output row.
    if (t < 32) {
        float b = wmin[0][t];
        int  bi = widx[0][t];
#pragma unroll
        for (int w = 1; w < 8; ++w) {
            float od = wmin[w][t];
            int   oi = widx[w][t];
            if (od < b || (od == b && oi < bi)) { b = od; bi = oi; }
        }
        int row = row0 + t;
        out_d[row] = sqrtf(fmaxf(b, EPSQ));   // sqrt deferred to the end
        out_i[2 * row]     = row;
        out_i[2 * row + 1] = bi;
    }
}

extern "C" void kernel_launch(void* const* d_in, const int* in_sizes, int n_in,
                              void* d_out, int out_size, void* d_ws, size_t ws_size,
                              hipStream_t stream) {
    const float* d1 = (const float*)d_in[0];
    const float* d2 = (const float*)d_in[1];
    const int B1 = in_sizes[0] / D;
    const int B2 = in_sizes[1] / D;

    float* outd = (float*)d_out;
    int*   outi = (int*)((float*)d_out + B1);

    float* sq  = (float*)d_ws;       // B1 norms then B2 norms
    float* sq1 = sq;
    float* sq2 = sq + B1;

    {
        int total = B1 + B2;
        int blocks = (total + 255) / 256;
        dm_norms_kernel<<<blocks, 256, 0, stream>>>(d1, d2, sq, B1, B2);
    }
    {
        int blocks = B1 / 32;
        dm_match_kernel<<<blocks, 256, 0, stream>>>(d1, d2, sq1, sq2, outd, outi, B2);
    }
}